// CVSS_Block_51479478010023
// MI455X (gfx1250) — compile-verified
//
#include <hip/hip_runtime.h>
#include <hip/hip_bf16.h>
#include <math.h>

typedef __attribute__((ext_vector_type(2))) float v2f;
typedef __attribute__((ext_vector_type(4))) float v4f;
typedef __attribute__((ext_vector_type(8))) float v8f;

constexpr int Bb  = 2;
constexpr int Hd  = 96;
constexpr int Di  = 192;
constexpr int Nst = 16;
constexpr int Rr  = 6;
constexpr int Hh  = 64;
constexpr int Wd  = 64;
constexpr int Lp  = Hh * Wd;        // 4096 == compile-time GEMM column count
constexpr int DTN = Rr + 2 * Nst;   // 38
constexpr float EPSV = 1e-5f;

__device__ __forceinline__ float sigmoidf_(float x) { return 1.0f / (1.0f + __expf(-x)); }
__device__ __forceinline__ float softplusf_(float x) {
    return fmaxf(x, 0.0f) + log1pf(__expf(-fabsf(x)));
}

// ---------------------------------------------------------------------------
// f32 WMMA GEMM:  O_z[M, Lp] = W_z[M, KR] @ X_z[KR, Lp]  (+ epilogue)
//   KP: K padded to multiple of 4 (fully unrolled), KR: real K,
//   MG: row guard for M%16!=0, EPI: 0 plain / 1 softplus+bias / 2 +skip*ss,
//   NT: N-tiles (16 cols each) per wave, A fragment reused across tiles.
// Column stride is the compile-time constant Lp, and per-lane base pointers
// are hoisted, so every load/store in the unrolled loop is base + immediate
// offset (fits 24-bit IOFFSET): the loop body is pure vmem + v_wmma.
// ---------------------------------------------------------------------------
template<int KP, int KR, bool MG, int EPI, int NT>
__global__ void gemm_f32_wmma(const float* __restrict__ Wt, const float* __restrict__ X,
                              float* __restrict__ O, int M,
                              int wmod, long zW, long zX, long zO,
                              const float* __restrict__ bias, long zBias,
                              const float* __restrict__ skip, const float* __restrict__ ss)
{
    const int z = blockIdx.z;
    Wt += (long)(z % wmod) * zW;
    X  += (long)z * zX;
    O  += (long)z * zO;

    const int wid  = threadIdx.x >> 5;
    const int lane = threadIdx.x & 31;
    const int n0 = (blockIdx.x * (blockDim.x >> 5) + wid) * (16 * NT);
    const int m0 = blockIdx.y * 16;

    const int r  = lane & 15;        // A row / B col / C col (within tile)
    const int kh = (lane >> 4) << 1; // 0 or 2

    int rowA = m0 + r;
    if (MG) rowA = (rowA < M) ? rowA : (M - 1);
    const float* Wr = Wt + (long)rowA * KR + kh;     // a.x=Wr[k0], a.y=Wr[k0+1]
    const float* Xp = X + (long)kh * Lp + n0 + r;    // b=Xp[k0*Lp + 16*t]

    v8f acc[NT];
    #pragma unroll
    for (int t = 0; t < NT; ++t) acc[t] = v8f{};

    #pragma unroll
    for (int k0 = 0; k0 < KP; k0 += 4) {
        v2f a;
        if (k0 + 4 <= KR) {                       // folds at compile time
            a.x = Wr[k0];
            a.y = Wr[k0 + 1];
            #pragma unroll
            for (int t = 0; t < NT; ++t) {
                v2f b;
                b.x = Xp[(long)k0 * Lp + 16 * t];
                b.y = Xp[(long)(k0 + 1) * Lp + 16 * t];
                acc[t] = __builtin_amdgcn_wmma_f32_16x16x4_f32(
                    false, a, false, b, (short)0, acc[t], false, false);
            }
        } else {                                   // K tail: clamp addr, zero vals
            const int ka  = k0 + kh;
            const int ka0 = (ka     < KR) ? ka       : (KR - 1);
            const int ka1 = (ka + 1 < KR) ? (ka + 1) : (KR - 1);
            a.x = (ka     < KR) ? Wt[(long)rowA * KR + ka0] : 0.0f;
            a.y = (ka + 1 < KR) ? Wt[(long)rowA * KR + ka1] : 0.0f;
            const float* Xt0 = X + (long)ka0 * Lp + n0 + r;
            const float* Xt1 = X + (long)ka1 * Lp + n0 + r;
            #pragma unroll
            for (int t = 0; t < NT; ++t) {
                v2f b;
                b.x = Xt0[16 * t];
                b.y = Xt1[16 * t];
                acc[t] = __builtin_amdgcn_wmma_f32_16x16x4_f32(
                    false, a, false, b, (short)0, acc[t], false, false);
            }
        }
    }

    const float ssv = (EPI == 2) ? ss[0] : 0.0f;
    const float* bz = (EPI == 1) ? (bias + (long)(z % wmod) * zBias) : nullptr;

    const int rbase = m0 + (lane >> 4) * 8;
    float* Ob = O + (long)rbase * Lp + n0 + r;                       // + v*Lp + 16*t
    const float* Sb = (EPI == 2) ? (skip + (long)z * zO + (long)rbase * Lp + n0 + r)
                                 : nullptr;
    #pragma unroll
    for (int v = 0; v < 8; ++v) {
        const int row = rbase + v;
        if (!MG || row < M) {
            const float bval = (EPI == 1) ? bz[row] : 0.0f;
            #pragma unroll
            for (int t = 0; t < NT; ++t) {
                float val = acc[t][v];
                if (EPI == 1)      val = softplusf_(val + bval);
                else if (EPI == 2) val += Sb[(long)v * Lp + 16 * t] * ssv;
                Ob[(long)v * Lp + 16 * t] = val;
            }
        }
    }
}

// ---------------------------------------------------------------------------
// Channel LayerNorm: one wave32 per pixel, reduce over C channels.
// ---------------------------------------------------------------------------
__global__ void chan_ln_k(const float* __restrict__ xin, const float* __restrict__ g,
                          const float* __restrict__ bv, float* __restrict__ xout, int C)
{
    const int gw   = blockIdx.x * (blockDim.x >> 5) + (threadIdx.x >> 5);
    const int lane = threadIdx.x & 31;
    if (gw >= Bb * Lp) return;
    const int b = gw / Lp, l = gw - b * Lp;

    const float* xp = xin + (long)b * C * Lp + l;
    float s = 0.0f, s2 = 0.0f;
    for (int c = lane; c < C; c += 32) {
        const float v = xp[(long)c * Lp];
        s += v; s2 += v * v;
    }
    #pragma unroll
    for (int o = 16; o >= 1; o >>= 1) { s += __shfl_xor(s, o); s2 += __shfl_xor(s2, o); }

    const float mu  = s / (float)C;
    const float var = fmaxf(s2 / (float)C - mu * mu, 0.0f);
    const float inv = rsqrtf(var + EPSV);

    float* op = xout + (long)b * C * Lp + l;
    for (int c = lane; c < C; c += 32) {
        const float v = xp[(long)c * Lp];
        op[(long)c * Lp] = (v - mu) * inv * g[c] + bv[c];
    }
}

// ---------------------------------------------------------------------------
// Depthwise 3x3 SAME conv + bias + SiLU
// ---------------------------------------------------------------------------
__global__ void dwconv_silu_k(const float* __restrict__ hin, const float* __restrict__ wgt,
                              const float* __restrict__ bias, float* __restrict__ out)
{
    const int idx = blockIdx.x * blockDim.x + threadIdx.x;
    if (idx >= Bb * Di * Lp) return;
    const int l = idx % Lp;
    const int c = (idx / Lp) % Di;
    const int b = idx / (Lp * Di);
    const int h = l / Wd, w = l % Wd;

    const float* wp = wgt + c * 9;
    const float* ip = hin + ((long)b * Di + c) * Lp;
    float acc = bias[c];
    #pragma unroll
    for (int dy = -1; dy <= 1; ++dy)
        #pragma unroll
        for (int dx = -1; dx <= 1; ++dx) {
            const int hh = h + dy, ww = w + dx;
            if (hh >= 0 && hh < Hh && ww >= 0 && ww < Wd)
                acc += wp[(dy + 1) * 3 + (dx + 1)] * ip[hh * Wd + ww];
        }
    out[idx] = acc * sigmoidf_(acc);
}

// ---------------------------------------------------------------------------
// Build the 4 directional sequences xs[b,k,c,:] from u[b,c,:]
// ---------------------------------------------------------------------------
__global__ void build_xs_k(const float* __restrict__ u, float* __restrict__ xs)
{
    const int idx = blockIdx.x * blockDim.x + threadIdx.x;
    if (idx >= Bb * Di * Lp) return;
    const int l = idx % Lp;
    const int c = (idx / Lp) % Di;
    const int b = idx / (Lp * Di);
    const int h = l / Wd, w = l % Wd;
    const int lt = w * Hh + h;

    const float v = u[idx];
    const long kstride = (long)Di * Lp;
    float* base = xs + (long)b * 4 * Di * Lp + (long)c * Lp;
    base[l] = v;
    base[kstride + lt] = v;
    base[2 * kstride + (Lp - 1 - l)] = v;
    base[3 * kstride + (Lp - 1 - lt)] = v;
}

// ---------------------------------------------------------------------------
// Selective scan: 16 lanes per chain (state dim N=16), 2 chains per wave32.
// 4 timesteps per b128 load/store; dependent chain: exp -> fma -> shfl reduce.
// ---------------------------------------------------------------------------
__global__ void sscan_k(const float* __restrict__ dts, const float* __restrict__ xdbl,
                        const float* __restrict__ xs, const float* __restrict__ A_log,
                        const float* __restrict__ Dsv, float* __restrict__ ys)
{
    const int chain = blockIdx.x * (blockDim.x >> 4) + (threadIdx.x >> 4);
    const int n = threadIdx.x & 15;
    if (chain >= Bb * 4 * Di) return;
    const int c = chain % Di;
    const int k = (chain / Di) % 4;
    const int b = chain / (4 * Di);

    const float An = -__expf(A_log[((long)k * Di + c) * Nst + n]);
    const float Dv = Dsv[k * Di + c];

    const float* dtp = dts  + ((long)(b * 4 + k) * Di  + c) * Lp;
    const float* up  = xs   + ((long)(b * 4 + k) * Di  + c) * Lp;
    const float* Bp  = xdbl + ((long)(b * 4 + k) * DTN + Rr + n) * Lp;
    const float* Cp  = xdbl + ((long)(b * 4 + k) * DTN + Rr + Nst + n) * Lp;
    float* yp        = ys   + ((long)(b * 4 + k) * Di  + c) * Lp;

    float hst = 0.0f;
    for (int t0 = 0; t0 < Lp; t0 += 4) {
        const v4f dt4 = *(const v4f*)(dtp + t0);
        const v4f u4  = *(const v4f*)(up + t0);
        const v4f B4  = *(const v4f*)(Bp + t0);
        const v4f C4  = *(const v4f*)(Cp + t0);
        v4f yout;
        #pragma unroll
        for (int j = 0; j < 4; ++j) {
            const float dt = dt4[j];
            const float ut = u4[j];
            hst = hst * __expf(dt * An) + (dt * ut) * B4[j];
            float y = hst * C4[j];
            #pragma unroll
            for (int o = 8; o >= 1; o >>= 1) y += __shfl_xor(y, o, 16);
            yout[j] = y + Dv * ut;
        }
        if (n == 0) *(v4f*)(yp + t0) = yout;
    }
}

// ---------------------------------------------------------------------------
// Merge 4 directional scan outputs back to spatial order
// ---------------------------------------------------------------------------
__global__ void combine_k(const float* __restrict__ ys, float* __restrict__ yc)
{
    const int idx = blockIdx.x * blockDim.x + threadIdx.x;
    if (idx >= Bb * Di * Lp) return;
    const int l = idx % Lp;
    const int c = (idx / Lp) % Di;
    const int b = idx / (Lp * Di);
    const int h = l / Wd, w = l % Wd;
    const int lt = w * Hh + h;

    const long kstride = (long)Di * Lp;
    const float* base = ys + (long)b * 4 * Di * Lp + (long)c * Lp;
    yc[idx] = base[l]
            + base[kstride + lt]
            + base[2 * kstride + (Lp - 1 - l)]
            + base[3 * kstride + (Lp - 1 - lt)];
}

// ---------------------------------------------------------------------------
extern "C" void kernel_launch(void* const* d_in, const int* in_sizes, int n_in,
                              void* d_out, int out_size, void* d_ws, size_t ws_size,
                              hipStream_t stream)
{
    (void)in_sizes; (void)n_in; (void)out_size; (void)ws_size;

    const float* x          = (const float*)d_in[0];
    const float* in_conv_w  = (const float*)d_in[1];
    const float* ln1_g      = (const float*)d_in[2];
    const float* ln1_b      = (const float*)d_in[3];
    const float* in_proj_w  = (const float*)d_in[4];
    const float* dw_w       = (const float*)d_in[5];
    const float* dw_b       = (const float*)d_in[6];
    const float* x_proj_w   = (const float*)d_in[7];
    const float* dt_proj_w  = (const float*)d_in[8];
    const float* dt_proj_b  = (const float*)d_in[9];
    const float* A_log      = (const float*)d_in[10];
    const float* Ds         = (const float*)d_in[11];
    const float* out_norm_g = (const float*)d_in[12];
    const float* out_norm_b = (const float*)d_in[13];
    const float* out_proj_w = (const float*)d_in[14];
    const float* skip_scale = (const float*)d_in[15];
    float* out = (float*)d_out;

    // workspace carve (floats)
    float* ws = (float*)d_ws;
    size_t o = 0;
    float* x1  = ws + o; o += (size_t)Bb * Hd * Lp;
    float* x1n = ws + o; o += (size_t)Bb * Hd * Lp;          // post-LN (= skip)
    float* hb  = ws + o; o += (size_t)Bb * Di * Lp;
    float* uu  = ws + o; o += (size_t)Bb * Di * Lp;
    float* xsb = ws + o; o += (size_t)Bb * 4 * Di * Lp;
    float* xdb = ws + o; o += (size_t)Bb * 4 * DTN * Lp;     // (dt rows | B | C)
    float* dtp = ws + o; o += (size_t)Bb * 4 * Di * Lp;
    float* ysb = ws + o; o += (size_t)Bb * 4 * Di * Lp;
    float* ycb = ws + o; o += (size_t)Bb * Di * Lp;
    float* yln = ws + o; o += (size_t)Bb * Di * Lp;

    const dim3 blkG(128);                 // 4 waves x 64 cols = 256 cols per block
    const int  gx = Lp / 256;             // 16
    const dim3 blkE(256);
    const int  nElemDi = Bb * Di * Lp;

    // 1) x1 = in_conv_w @ x            (M=96, K=96) per batch
    gemm_f32_wmma<96, 96, false, 0, 4><<<dim3(gx, Hd / 16, Bb), blkG, 0, stream>>>(
        in_conv_w, x, x1, Hd,
        1, 0, (long)Hd * Lp, (long)Hd * Lp, nullptr, 0, nullptr, nullptr);

    // 2) channel LN -> x1n (skip)
    chan_ln_k<<<dim3((Bb * Lp) / 8), blkE, 0, stream>>>(x1, ln1_g, ln1_b, x1n, Hd);

    // 3) hb = in_proj_w @ x1n          (M=192, K=96) per batch
    gemm_f32_wmma<96, 96, false, 0, 4><<<dim3(gx, Di / 16, Bb), blkG, 0, stream>>>(
        in_proj_w, x1n, hb, Di,
        1, 0, (long)Hd * Lp, (long)Di * Lp, nullptr, 0, nullptr, nullptr);

    // 4) depthwise 3x3 + bias + SiLU -> uu
    dwconv_silu_k<<<dim3(nElemDi / 256), blkE, 0, stream>>>(hb, dw_w, dw_b, uu);

    // 5) build 4 directional sequences
    build_xs_k<<<dim3(nElemDi / 256), blkE, 0, stream>>>(uu, xsb);

    // 6) xdb = x_proj_w[k] @ xs[b,k]   (M=38, K=192) per (b,k)
    gemm_f32_wmma<192, 192, true, 0, 4><<<dim3(gx, (DTN + 15) / 16, Bb * 4), blkG, 0, stream>>>(
        x_proj_w, xsb, xdb, DTN,
        4, (long)DTN * Di, (long)Di * Lp, (long)DTN * Lp, nullptr, 0, nullptr, nullptr);

    // 7) dtp = softplus(dt_proj_w[k] @ xdb[b,k][0:6] + dt_proj_b[k])  (M=192, K=6->8)
    gemm_f32_wmma<8, 6, false, 1, 4><<<dim3(gx, Di / 16, Bb * 4), blkG, 0, stream>>>(
        dt_proj_w, xdb, dtp, Di,
        4, (long)Di * Rr, (long)DTN * Lp, (long)Di * Lp,
        dt_proj_b, (long)Di, nullptr, nullptr);

    // 8) selective scan: 1536 chains, 16 lanes/chain
    sscan_k<<<dim3((Bb * 4 * Di) / 16), blkE, 0, stream>>>(dtp, xdb, xsb, A_log, Ds, ysb);

    // 9) merge 4 directions
    combine_k<<<dim3(nElemDi / 256), blkE, 0, stream>>>(ysb, ycb);

    // 10) out_norm channel LN
    chan_ln_k<<<dim3((Bb * Lp) / 8), blkE, 0, stream>>>(ycb, out_norm_g, out_norm_b, yln, Di);

    // 11) out = out_proj_w @ yln + skip * skip_scale   (M=96, K=192)
    gemm_f32_wmma<192, 192, false, 2, 4><<<dim3(gx, Hd / 16, Bb), blkG, 0, stream>>>(
        out_proj_w, yln, out, Hd,
        1, 0, (long)Di * Lp, (long)Hd * Lp,
        nullptr, 0, x1n, skip_scale);
}